// GaussianSplattingRender_55542517072266
// MI455X (gfx1250) — compile-verified
//
#include <hip/hip_runtime.h>
#include <hip/hip_bf16.h>

typedef __attribute__((ext_vector_type(2))) float v2f;
typedef __attribute__((ext_vector_type(8))) float v8f;

// ---------------------------------------------------------------------------
// Kernel 1: per-gaussian preprocessing.
// Record layout (12 floats, 48 B, 16B-aligned):
//  [0]=u [1]=v [2]=ca [3]=cb [4]=cc [5]=depth [6..7]=pad [8]=opac [9..11]=rgb
// ---------------------------------------------------------------------------
__global__ void __launch_bounds__(256) gs_preprocess(
    const float* __restrict__ means, const float* __restrict__ quats,
    const float* __restrict__ scales, const float* __restrict__ opacities,
    const float* __restrict__ colors, const float* __restrict__ c2w,
    const float* __restrict__ Km, float* __restrict__ rec, int N)
{
    int i = blockIdx.x * blockDim.x + threadIdx.x;
    if (i >= N) return;

    // viewmat = rigid inverse of camtoworlds (4x4 row-major)
    float Rw[3][3], tw[3];
    #pragma unroll
    for (int r = 0; r < 3; ++r)
        #pragma unroll
        for (int c = 0; c < 3; ++c)
            Rw[r][c] = c2w[c * 4 + r];
    const float tc0 = c2w[3], tc1 = c2w[7], tc2 = c2w[11];
    #pragma unroll
    for (int r = 0; r < 3; ++r)
        tw[r] = -(Rw[r][0] * tc0 + Rw[r][1] * tc1 + Rw[r][2] * tc2);

    const float mx = means[3 * i + 0], my = means[3 * i + 1], mz = means[3 * i + 2];
    float t0 = Rw[0][0] * mx + Rw[0][1] * my + Rw[0][2] * mz + tw[0];
    float t1 = Rw[1][0] * mx + Rw[1][1] * my + Rw[1][2] * mz + tw[1];
    float t2 = Rw[2][0] * mx + Rw[2][1] * my + Rw[2][2] * mz + tw[2];
    const float depth = t2;
    const float tz = fmaxf(depth, 0.01f);

    const float fx = Km[0], fy = Km[4], cx = Km[2], cy = Km[5];
    const float u = fx * t0 / tz + cx;
    const float v = fy * t1 / tz + cy;

    // quat -> rotmat (normalized)
    float qw = quats[4 * i + 0], qx = quats[4 * i + 1], qy = quats[4 * i + 2], qz = quats[4 * i + 3];
    const float qn = rsqrtf(qw * qw + qx * qx + qy * qy + qz * qz);
    qw *= qn; qx *= qn; qy *= qn; qz *= qn;
    float Rq[3][3];
    Rq[0][0] = 1.f - 2.f * (qy * qy + qz * qz);
    Rq[0][1] = 2.f * (qx * qy - qw * qz);
    Rq[0][2] = 2.f * (qx * qz + qw * qy);
    Rq[1][0] = 2.f * (qx * qy + qw * qz);
    Rq[1][1] = 1.f - 2.f * (qx * qx + qz * qz);
    Rq[1][2] = 2.f * (qy * qz - qw * qx);
    Rq[2][0] = 2.f * (qx * qz - qw * qy);
    Rq[2][1] = 2.f * (qy * qz + qw * qx);
    Rq[2][2] = 1.f - 2.f * (qx * qx + qy * qy);

    const float s0 = __expf(scales[3 * i + 0]);
    const float s1 = __expf(scales[3 * i + 1]);
    const float s2 = __expf(scales[3 * i + 2]);

    // M = Rq * diag(s);  cov3d = M M^T
    float M[3][3];
    #pragma unroll
    for (int r = 0; r < 3; ++r) { M[r][0] = Rq[r][0] * s0; M[r][1] = Rq[r][1] * s1; M[r][2] = Rq[r][2] * s2; }
    float c3[3][3];
    #pragma unroll
    for (int r = 0; r < 3; ++r)
        #pragma unroll
        for (int c = 0; c < 3; ++c)
            c3[r][c] = M[r][0] * M[c][0] + M[r][1] * M[c][1] + M[r][2] * M[c][2];

    // covc = Rw cov3d Rw^T
    float tmp[3][3], cvc[3][3];
    #pragma unroll
    for (int r = 0; r < 3; ++r)
        #pragma unroll
        for (int c = 0; c < 3; ++c)
            tmp[r][c] = Rw[r][0] * c3[0][c] + Rw[r][1] * c3[1][c] + Rw[r][2] * c3[2][c];
    #pragma unroll
    for (int r = 0; r < 3; ++r)
        #pragma unroll
        for (int c = 0; c < 3; ++c)
            cvc[r][c] = tmp[r][0] * Rw[c][0] + tmp[r][1] * Rw[c][1] + tmp[r][2] * Rw[c][2];

    // J (2x3), cov2d = J covc J^T
    const float z1 = 1.0f / tz;
    const float J00 = fx * z1, J02 = -fx * t0 * z1 * z1;
    const float J11 = fy * z1, J12 = -fy * t1 * z1 * z1;
    const float w00 = J00 * cvc[0][0] + J02 * cvc[2][0];
    const float w01 = J00 * cvc[0][1] + J02 * cvc[2][1];
    const float w02 = J00 * cvc[0][2] + J02 * cvc[2][2];
    const float w11 = J11 * cvc[1][1] + J12 * cvc[2][1];
    const float w12 = J11 * cvc[1][2] + J12 * cvc[2][2];
    const float a = w00 * J00 + w02 * J02 + 0.3f;
    const float b = w01 * J11 + w02 * J12;
    const float c = w11 * J11 + w12 * J12 + 0.3f;
    const float det = a * c - b * b;
    const float ca = c / det, cb = -b / det, cc = a / det;

    const float op = 1.0f / (1.0f + __expf(-opacities[i]));

    // SH-3 color
    float dx = mx - tc0, dy = my - tc1, dz = mz - tc2;
    const float dn = rsqrtf(dx * dx + dy * dy + dz * dz);
    dx *= dn; dy *= dn; dz *= dn;
    const float xx = dx * dx, yy = dy * dy, zz = dz * dz;
    const float xy = dx * dy, yz = dy * dz, xz = dx * dz;
    float bas[16];
    bas[0]  = 0.28209479177387814f;
    bas[1]  = -0.4886025119029199f * dy;
    bas[2]  =  0.4886025119029199f * dz;
    bas[3]  = -0.4886025119029199f * dx;
    bas[4]  =  1.0925484305920792f * xy;
    bas[5]  = -1.0925484305920792f * yz;
    bas[6]  =  0.31539156525252005f * (2.f * zz - xx - yy);
    bas[7]  = -1.0925484305920792f * xz;
    bas[8]  =  0.5462742152960396f * (xx - yy);
    bas[9]  = -0.5900435899266435f * dy * (3.f * xx - yy);
    bas[10] =  2.890611442640554f * xy * dz;
    bas[11] = -0.4570457994644658f * dy * (4.f * zz - xx - yy);
    bas[12] =  0.3731763325901154f * dz * (2.f * zz - 3.f * xx - 3.f * yy);
    bas[13] = -0.4570457994644658f * dx * (4.f * zz - xx - yy);
    bas[14] =  1.445305721320277f * dz * (xx - yy);
    bas[15] = -0.5900435899266435f * dx * (xx - 3.f * yy);
    float rgb[3];
    #pragma unroll
    for (int chn = 0; chn < 3; ++chn) {
        float acc = 0.f;
        #pragma unroll
        for (int k = 0; k < 16; ++k)
            acc += bas[k] * colors[(i * 16 + k) * 3 + chn];
        rgb[chn] = fmaxf(acc + 0.5f, 0.0f);
    }

    float* R = rec + (size_t)i * 12;
    R[0] = u;  R[1] = v;  R[2] = ca; R[3] = cb;
    R[4] = cc; R[5] = depth; R[6] = 0.f; R[7] = 0.f;
    R[8] = op; R[9] = rgb[0]; R[10] = rgb[1]; R[11] = rgb[2];
}

// ---------------------------------------------------------------------------
// Kernel 2: stable rank sort by depth (O(N^2), N=2048 -> trivial) + scatter.
// ---------------------------------------------------------------------------
__global__ void __launch_bounds__(256) gs_sort_scatter(
    const float* __restrict__ rec, float* __restrict__ srec, int N)
{
    int i = blockIdx.x * blockDim.x + threadIdx.x;
    if (i >= N) return;
    const float di = rec[(size_t)i * 12 + 5];
    int rank = 0;
    for (int j = 0; j < N; ++j) {
        const float dj = rec[(size_t)j * 12 + 5];
        rank += (dj < di) || (dj == di && j < i);
    }
    #pragma unroll
    for (int k = 0; k < 12; ++k)
        srec[(size_t)rank * 12 + k] = rec[(size_t)i * 12 + k];
}

// ---------------------------------------------------------------------------
// Kernel 3: compositing. One wave32 = 16 pixels.
//  - Whole sorted record array (N*48 B; 96 KB for N=2048) staged into LDS once
//    per block via async global->LDS copies (ASYNCcnt path), shared by 8 waves.
//  - Per 16-gaussian chunk: sigma(16g x 16px) via rank-6 factorization ->
//    2x V_WMMA_F32_16X16X4_F32, then half-lane local composite + associative
//    (T,C) merge via shfl_xor(16).
// ---------------------------------------------------------------------------
__global__ void __launch_bounds__(256) gs_render(
    const float* __restrict__ srec, float* __restrict__ out, int N, int W, int H)
{
    extern __shared__ __align__(16) float smem[];   // N*12 floats

    // ---- Cooperative async stage: global -> LDS (16 B per issue) ----------
    const int nvec = (N * 12) >> 2;                 // # of float4 packets
    for (int idx = threadIdx.x; idx < nvec; idx += blockDim.x) {
        unsigned lds_addr = (unsigned)(uintptr_t)(&smem[idx * 4]);
        unsigned long long gaddr =
            (unsigned long long)(uintptr_t)(srec + (size_t)idx * 4);
        asm volatile("global_load_async_to_lds_b128 %0, %1, off"
                     :: "v"(lds_addr), "v"(gaddr) : "memory");
    }
    asm volatile("s_wait_asynccnt 0x0" ::: "memory");
    __syncthreads();

    const int tid  = threadIdx.x;
    const int lane = tid & 31;
    const int n    = lane & 15;          // pixel slot within the wave's strip
    const bool hi  = lane >= 16;
    const int gw   = blockIdx.x * (blockDim.x >> 5) + (tid >> 5);
    const int p    = gw * 16 + n;        // global pixel id (row-major)
    const float gx = (float)(p % W) + 0.5f;
    const float gy = (float)(p / W) + 0.5f;

    // B matrix (8 x 16 over two K-groups), constant over the gaussian loop.
    // K rows: [gx^2, gy^2, gx*gy, gx | gy, 1, 0, 0]
    v2f b0, b1;
    if (!hi) { b0.x = gx * gx; b0.y = gy * gy; b1.x = gy;  b1.y = 1.0f; }
    else     { b0.x = gx * gy; b0.y = gx;      b1.x = 0.f; b1.y = 0.f;  }

    float T = 1.0f, Cr = 0.f, Cg = 0.f, Cb = 0.f;  // duplicated in lane pair

    const int nchunks = N >> 4;
    for (int chk = 0; chk < nchunks; ++chk) {
        const int gbase = chk << 4;
        const int g = gbase + n;
        const float4 uvab = *(const float4*)(&smem[g * 12]);   // u,v,ca,cb
        const float  ccv  = smem[g * 12 + 4];                  // cc
        const float u_ = uvab.x, v_ = uvab.y, ca = uvab.z, cb = uvab.w, cc = ccv;

        // A rows (16 x 8 over two K-groups), per-gaussian coefficients:
        // [ca/2, cc/2, cb, -(ca*u+cb*v) | -(cc*v+cb*u), q, 0, 0]
        v2f a0, a1;
        if (!hi) {
            a0.x = 0.5f * ca; a0.y = 0.5f * cc;
            a1.x = -(cc * v_ + cb * u_);
            a1.y = 0.5f * ca * u_ * u_ + 0.5f * cc * v_ * v_ + cb * u_ * v_;
        } else {
            a0.x = cb;        a0.y = -(ca * u_ + cb * v_);
            a1.x = 0.f;       a1.y = 0.f;
        }

        v8f sig = {};
        sig = __builtin_amdgcn_wmma_f32_16x16x4_f32(false, a0, false, b0,
                                                    (short)0, sig, false, false);
        sig = __builtin_amdgcn_wmma_f32_16x16x4_f32(false, a1, false, b1,
                                                    (short)0, sig, false, false);

        // Local composite: lanes 0-15 own chunk rows 0-7, lanes 16-31 rows 8-15.
        float Tl = 1.0f, Lr = 0.f, Lg = 0.f, Lb = 0.f;
        const int rowbase = gbase + (hi ? 8 : 0);
        #pragma unroll
        for (int r = 0; r < 8; ++r) {
            const int j = rowbase + r;
            const float4 orgb = *(const float4*)(&smem[j * 12 + 8]); // opac,rgb
            const float dj    = smem[j * 12 + 5];
            const float s     = sig[r];
            float al = orgb.x * __expf(-fmaxf(s, 0.0f));
            const bool vis = (dj > 0.01f) && (al >= (1.0f / 255.0f));
            al = vis ? fminf(al, 0.999f) : 0.0f;
            const float w = al * Tl;
            Lr += w * orgb.y; Lg += w * orgb.z; Lb += w * orgb.w;
            Tl *= (1.0f - al);
        }

        // Merge lane pair with associative compositing operator.
        const float To = __shfl_xor(Tl, 16, 32);
        const float Ro = __shfl_xor(Lr, 16, 32);
        const float Go = __shfl_xor(Lg, 16, 32);
        const float Bo = __shfl_xor(Lb, 16, 32);
        float Tf, Ts, Rf, Gf, Bf, Rs, Gs, Bs;
        if (!hi) { Tf = Tl; Rf = Lr; Gf = Lg; Bf = Lb; Ts = To; Rs = Ro; Gs = Go; Bs = Bo; }
        else     { Tf = To; Rf = Ro; Gf = Go; Bf = Bo; Ts = Tl; Rs = Lr; Gs = Lg; Bs = Lb; }
        const float Rc = Rf + Tf * Rs;
        const float Gc = Gf + Tf * Gs;
        const float Bc = Bf + Tf * Bs;
        const float Tc = Tf * Ts;
        Cr += T * Rc; Cg += T * Gc; Cb += T * Bc;
        T *= Tc;
    }

    if (!hi) {
        out[3 * p + 0] = Cr;
        out[3 * p + 1] = Cg;
        out[3 * p + 2] = Cb;
        out[(size_t)W * H * 3 + p] = 1.0f - T;
    }
}

// ---------------------------------------------------------------------------
extern "C" void kernel_launch(void* const* d_in, const int* in_sizes, int n_in,
                              void* d_out, int out_size, void* d_ws, size_t ws_size,
                              hipStream_t stream) {
    const float* means     = (const float*)d_in[0];
    const float* quats     = (const float*)d_in[1];
    const float* scales    = (const float*)d_in[2];
    const float* opacities = (const float*)d_in[3];
    const float* colors    = (const float*)d_in[4];
    const float* c2w       = (const float*)d_in[5];
    const float* Km        = (const float*)d_in[6];
    (void)n_in; (void)out_size; (void)ws_size;

    const int N = in_sizes[0] / 3;     // 2048
    const int W = 128, H = 128;        // reference setup

    float* rec  = (float*)d_ws;                      // N*12 floats
    float* srec = rec + (size_t)N * 12;              // N*12 floats (sorted)

    gs_preprocess<<<(N + 255) / 256, 256, 0, stream>>>(
        means, quats, scales, opacities, colors, c2w, Km, rec, N);
    gs_sort_scatter<<<(N + 255) / 256, 256, 0, stream>>>(rec, srec, N);

    const int waves  = (W * H) / 16;   // 1024 waves, 16 px each
    const int blocks = waves / 8;      // 256-thread blocks = 8 wave32s
    const size_t ldsBytes = (size_t)N * 12 * sizeof(float);   // 96 KB @ N=2048
    gs_render<<<blocks, 256, ldsBytes, stream>>>(srec, (float*)d_out, N, W, H);
}